// DQN2_9225589751878
// MI455X (gfx1250) — compile-verified
//
#include <hip/hip_runtime.h>

typedef __attribute__((ext_vector_type(2))) float v2f;
typedef __attribute__((ext_vector_type(8))) float v8f;

namespace {
constexpr int NU   = 256;
constexpr int NI   = 512;
constexpr int NJ   = 31;
constexpr int EIN  = 3;
constexpr int NEMB = 10;
constexpr int NFIT = 32;
constexpr int KSL  = 2;             // N_EMB // 4
constexpr int DDIM = KSL * NEMB;    // 20
constexpr int PAIRS = NU * NI;      // 131072
constexpr int THREADS = 256;
constexpr int WAVES = THREADS / 32;
constexpr int PAIRS_PER_BLOCK = THREADS;  // one pair per lane-slot
} // namespace

// Wave-local LDS ordering: a wave's DS ops execute in order; after DScnt==0
// all prior LDS stores from this wave are visible to all its lanes. The
// "memory" clobber stops the compiler from moving LDS accesses across it.
__device__ __forceinline__ void wave_lds_fence() {
    asm volatile("s_wait_dscnt 0x0" ::: "memory");
}

__global__ __launch_bounds__(THREADS) void dqn2_fused_kernel(
    const float* __restrict__ Sg,  const float* __restrict__ Rm,
    const float* __restrict__ We1, const float* __restrict__ be1,
    const float* __restrict__ We2, const float* __restrict__ be2,
    const float* __restrict__ Wf1, const float* __restrict__ bf1,
    const float* __restrict__ Wf2, const float* __restrict__ bf2,
    const float* __restrict__ Wf3, const float* __restrict__ bf3,
    float* __restrict__ out)
{
    // ---- LDS: all weights staged once per block + per-wave WMMA staging ----
    __shared__ float sWe1[EIN * NEMB];
    __shared__ float sbe1[NEMB];
    __shared__ float sWe2[NEMB * NEMB];
    __shared__ float sbe2[NEMB];
    __shared__ float sWf1[DDIM * NFIT];
    __shared__ float sbf1[NFIT];
    __shared__ float sWf2[NFIT * NFIT];
    __shared__ float sbf2[NFIT];
    __shared__ float sWf3[NFIT];
    __shared__ float sbf3;
    __shared__ float sD[WAVES][32][DDIM + 1];   // descriptor rows, stride 21
    __shared__ float sF[WAVES][16][NFIT + 1];   // activation transpose, stride 33

    const int tid = threadIdx.x;
    for (int i = tid; i < EIN * NEMB;  i += THREADS) sWe1[i] = We1[i];
    for (int i = tid; i < NEMB;        i += THREADS) sbe1[i] = be1[i];
    for (int i = tid; i < NEMB * NEMB; i += THREADS) sWe2[i] = We2[i];
    for (int i = tid; i < NEMB;        i += THREADS) sbe2[i] = be2[i];
    for (int i = tid; i < DDIM * NFIT; i += THREADS) sWf1[i] = Wf1[i];
    for (int i = tid; i < NFIT;        i += THREADS) sbf1[i] = bf1[i];
    for (int i = tid; i < NFIT * NFIT; i += THREADS) sWf2[i] = Wf2[i];
    for (int i = tid; i < NFIT;        i += THREADS) sbf2[i] = bf2[i];
    for (int i = tid; i < NFIT;        i += THREADS) sWf3[i] = Wf3[i];
    if (tid == 0) sbf3 = bf3[0];
    __syncthreads();   // the only block-wide barrier: weights are read-only after this

    const int wave = tid >> 5;
    const int lane = tid & 31;
    const int pairBase = blockIdx.x * PAIRS_PER_BLOCK + wave * 32;
    const int p = pairBase + lane;   // this lane's (u,i) pair

    // ================= Phase 1: embedding MLP + descriptor (per-lane) ======
    float T1[KSL][4] = {};
    float T2[4][NEMB] = {};
    {
        const float*  sg = Sg + (size_t)p * NJ * EIN;
        const float4* rp = reinterpret_cast<const float4*>(Rm + (size_t)p * NJ * 4);
        #pragma unroll 1
        for (int j = 0; j < NJ; ++j) {
            __builtin_prefetch(sg + 64, 0, 1);                // ~256B ahead
            __builtin_prefetch((const float*)(rp + j + 16), 0, 1);
            const float s0 = sg[0], s1 = sg[1], s2 = sg[2];
            const float4 r4 = rp[j];                          // global_load_b128
            sg += EIN;
            const float r[4] = { r4.x, r4.y, r4.z, r4.w };

            float h[NEMB];
            #pragma unroll
            for (int n = 0; n < NEMB; ++n) {
                float v = fmaf(s0, sWe1[n],
                          fmaf(s1, sWe1[NEMB + n],
                          fmaf(s2, sWe1[2 * NEMB + n], sbe1[n])));
                h[n] = v > 0.0f ? v : 0.0f;
            }
            float g[NEMB];
            #pragma unroll
            for (int n = 0; n < NEMB; ++n) {
                float v = sbe2[n];
                #pragma unroll
                for (int m = 0; m < NEMB; ++m) v = fmaf(h[m], sWe2[m * NEMB + n], v);
                g[n] = v;
            }
            #pragma unroll
            for (int l = 0; l < 4; ++l) {
                T1[0][l] = fmaf(g[0], r[l], T1[0][l]);
                T1[1][l] = fmaf(g[1], r[l], T1[1][l]);
                #pragma unroll
                for (int n = 0; n < NEMB; ++n)
                    T2[l][n] = fmaf(r[l], g[n], T2[l][n]);
            }
        }
    }
    // D = T1 @ T2  ->  [2,10] flattened to 20; park in LDS for WMMA A-layout reads
    #pragma unroll
    for (int k = 0; k < KSL; ++k)
        #pragma unroll
        for (int n = 0; n < NEMB; ++n) {
            float v = 0.0f;
            #pragma unroll
            for (int l = 0; l < 4; ++l) v = fmaf(T1[k][l], T2[l][n], v);
            sD[wave][lane][k * NEMB + n] = v;
        }
    wave_lds_fence();   // per-wave buffer: dscnt wait suffices, no block barrier

    // ================= Phase 2: fit MLP via V_WMMA_F32_16X16X4_F32 =========
    // A 16x4 layout: lanes 0-15 hold K=4c,4c+1 ; lanes 16-31 hold K=4c+2,4c+3
    // B 4x16 layout: vgpr0 = rows {4c, 4c+2} striped across lane halves, vgpr1 = {4c+1, 4c+3}
    // C/D layout   : vgpr r = element (M = r + 8*(lane>>4), N = lane&15 [+16*ntile])
    const int half = lane >> 4;
    const int mrow = lane & 15;
    const float bias1_0 = sbf1[mrow], bias1_1 = sbf1[mrow + 16];
    const float bias2_0 = sbf2[mrow], bias2_1 = sbf2[mrow + 16];
    const float b3 = sbf3;

    // Hoist all fit-layer B-operand fragments into registers (once per wave);
    // inside the WMMA loops only the A fragments touch LDS.
    v2f wf1A[5], wf1B[5];
    #pragma unroll
    for (int c = 0; c < 5; ++c) {
        const int k0 = 4 * c + 2 * half;
        wf1A[c][0] = sWf1[k0 * NFIT + mrow];
        wf1A[c][1] = sWf1[(k0 + 1) * NFIT + mrow];
        wf1B[c][0] = sWf1[k0 * NFIT + 16 + mrow];
        wf1B[c][1] = sWf1[(k0 + 1) * NFIT + 16 + mrow];
    }
    v2f wf2A[8], wf2B[8];
    #pragma unroll
    for (int c = 0; c < 8; ++c) {
        const int k0 = 4 * c + 2 * half;
        wf2A[c][0] = sWf2[k0 * NFIT + mrow];
        wf2A[c][1] = sWf2[(k0 + 1) * NFIT + mrow];
        wf2B[c][0] = sWf2[k0 * NFIT + 16 + mrow];
        wf2B[c][1] = sWf2[(k0 + 1) * NFIT + 16 + mrow];
    }

    #pragma unroll 1
    for (int T = 0; T < 2; ++T) {           // two 16-pair tiles per wave
        // ---- fit1: [16 x 20] @ [20 x 32] + b, ReLU ----
        v8f acc0, acc1;
        #pragma unroll
        for (int e = 0; e < 8; ++e) { acc0[e] = bias1_0; acc1[e] = bias1_1; }
        #pragma unroll
        for (int c = 0; c < 5; ++c) {       // K = 20 -> 5 chunks of 4
            const int k0 = 4 * c + 2 * half;
            v2f a;
            a[0] = sD[wave][T * 16 + mrow][k0];
            a[1] = sD[wave][T * 16 + mrow][k0 + 1];
            acc0 = __builtin_amdgcn_wmma_f32_16x16x4_f32(false, a, false, wf1A[c], (short)0, acc0, false, false);
            acc1 = __builtin_amdgcn_wmma_f32_16x16x4_f32(false, a, false, wf1B[c], (short)0, acc1, false, false);
        }
        #pragma unroll
        for (int r = 0; r < 8; ++r) {       // ReLU + transpose through LDS
            const float v0 = acc0[r], v1 = acc1[r];
            sF[wave][r + 8 * half][mrow]      = v0 > 0.0f ? v0 : 0.0f;
            sF[wave][r + 8 * half][mrow + 16] = v1 > 0.0f ? v1 : 0.0f;
        }
        wave_lds_fence();

        // ---- fit2: [16 x 32] @ [32 x 32] + b, ReLU ----
        v8f acc2, acc3;
        #pragma unroll
        for (int e = 0; e < 8; ++e) { acc2[e] = bias2_0; acc3[e] = bias2_1; }
        #pragma unroll
        for (int c = 0; c < 8; ++c) {       // K = 32 -> 8 chunks of 4
            const int k0 = 4 * c + 2 * half;
            v2f a;
            a[0] = sF[wave][mrow][k0];
            a[1] = sF[wave][mrow][k0 + 1];
            acc2 = __builtin_amdgcn_wmma_f32_16x16x4_f32(false, a, false, wf2A[c], (short)0, acc2, false, false);
            acc3 = __builtin_amdgcn_wmma_f32_16x16x4_f32(false, a, false, wf2B[c], (short)0, acc3, false, false);
        }
        wave_lds_fence();                   // fit2 A-reads done before overwrite
        #pragma unroll
        for (int r = 0; r < 8; ++r) {
            const float v0 = acc2[r], v1 = acc3[r];
            sF[wave][r + 8 * half][mrow]      = v0 > 0.0f ? v0 : 0.0f;
            sF[wave][r + 8 * half][mrow + 16] = v1 > 0.0f ? v1 : 0.0f;
        }
        wave_lds_fence();

        // ---- fit3: [16 x 32] @ [32 x 1] + b3 ----
        if (lane < 16) {
            float q = b3;
            #pragma unroll
            for (int k = 0; k < NFIT; ++k) q = fmaf(sF[wave][lane][k], sWf3[k], q);
            out[pairBase + T * 16 + lane] = q;
        }
        asm volatile("" ::: "memory");      // keep next tile's sF stores after reads
    }
}

extern "C" void kernel_launch(void* const* d_in, const int* in_sizes, int n_in,
                              void* d_out, int out_size, void* d_ws, size_t ws_size,
                              hipStream_t stream) {
    (void)in_sizes; (void)n_in; (void)out_size; (void)d_ws; (void)ws_size;
    const float* Sg  = (const float*)d_in[0];
    const float* Rm  = (const float*)d_in[1];
    const float* We1 = (const float*)d_in[2];
    const float* be1 = (const float*)d_in[3];
    const float* We2 = (const float*)d_in[4];
    const float* be2 = (const float*)d_in[5];
    const float* Wf1 = (const float*)d_in[6];
    const float* bf1 = (const float*)d_in[7];
    const float* Wf2 = (const float*)d_in[8];
    const float* bf2 = (const float*)d_in[9];
    const float* Wf3 = (const float*)d_in[10];
    const float* bf3 = (const float*)d_in[11];
    float* out = (float*)d_out;

    const int blocks = PAIRS / PAIRS_PER_BLOCK;   // 512
    dqn2_fused_kernel<<<dim3(blocks), dim3(THREADS), 0, stream>>>(
        Sg, Rm, We1, be1, We2, be2, Wf1, bf1, Wf2, bf2, Wf3, bf3, out);
}